// MAMConv2d_3470333575540
// MI455X (gfx1250) — compile-verified
//
#include <hip/hip_runtime.h>

// ---------------------------------------------------------------------------
// MAM-Conv2d for MI455X (gfx1250), compile-only tuning.
// - v_wmma_f32_16x16x4_f32 outer-product trick (K-slot 0 only) computes
//   prod[m,n] = patch[m,k] * w[n,k] exactly in f32 on the matrix pipe.
// - VALU v_max3/v_min3 accumulation co-executes with WMMA.
// - Weights double-buffered in LDS via async global->LDS loads; in-order
//   ASYNCcnt lets us wait on "<= 16 outstanding" so the next chunk's L2
//   fetch overlaps the current chunk's compute.
// ---------------------------------------------------------------------------

typedef __attribute__((ext_vector_type(2))) float v2f;
typedef __attribute__((ext_vector_type(8))) float v8f;

#define BB_   16
#define CIN   128
#define HH_   32
#define WW_   32
#define COUT  128
#define HO    30
#define WO    30

#define XS_COLS 18              // 16 pixels + (KW-1)
#define WCHUNK  32              // c-values per staged weight chunk
#define NCPT    (CIN / WCHUNK)  // 4 chunks per tap
#define NCHP    (9 * NCPT)      // 36 chunks total
#define STAGE_OPS (WCHUNK * COUT / 256)  // 16 async ops per thread per chunk

#define AS1 __attribute__((address_space(1)))
#define AS3 __attribute__((address_space(3)))

#if __has_builtin(__builtin_amdgcn_global_load_async_to_lds_b32)
#define HAVE_ASYNC_LDS 1
#else
#define HAVE_ASYNC_LDS 0
#endif

__device__ __forceinline__ void stage_g2l(const float* g, float* l) {
#if HAVE_ASYNC_LDS
  __builtin_amdgcn_global_load_async_to_lds_b32((AS1 int*)g, (AS3 int*)l,
                                                0, 0);
#else
  *l = *g;                      // plain ds_store staging fallback
#endif
}

template <int N>
__device__ __forceinline__ void wait_async() {
#if HAVE_ASYNC_LDS
#if __has_builtin(__builtin_amdgcn_s_wait_asynccnt)
  __builtin_amdgcn_s_wait_asynccnt(N);
#else
  asm volatile("s_wait_asynccnt %0" ::"i"(N) : "memory");
#endif
#endif
}

__device__ __forceinline__ v8f vmax8(v8f a, v8f b) {
  v8f r;
#pragma unroll
  for (int i = 0; i < 8; ++i) r[i] = __builtin_fmaxf(a[i], b[i]);
  return r;
}
__device__ __forceinline__ v8f vmin8(v8f a, v8f b) {
  v8f r;
#pragma unroll
  for (int i = 0; i < 8; ++i) r[i] = __builtin_fminf(a[i], b[i]);
  return r;
}

__global__ __launch_bounds__(256) void mamconv2d_kernel(
    const float* __restrict__ x,      // [B, CIN, H, W]
    const float* __restrict__ w,      // [COUT, CIN, 3, 3]
    const float* __restrict__ bias,   // [COUT]
    float* __restrict__ out)          // [B, COUT, HO, WO]
{
  __shared__ float xs[3 * CIN * XS_COLS];     // [i][c][col]        27.0 KB
  __shared__ float ws[2 * WCHUNK * COUT];     // ping-pong [cl][co] 32.0 KB

  const int tid  = threadIdx.x;
  const int lane = tid & 31;
  const int wave = tid >> 5;           // 0..7
  const int n0   = wave * 16;          // cout base of this wave's tile
  const int lan  = lane & 15;
  const bool loA = lane < 16;

  const int bid = blockIdx.x;          // B*HO*2 blocks
  const int wt  = bid & 1;
  const int ho  = (bid >> 1) % HO;
  const int b   = bid / (2 * HO);
  const int wo0 = wt * 16;

  // ---- stage x slice: xs[(i*CIN+c)*XS_COLS + col] = x[b, c, ho+i, wo0+col]
  for (int idx = tid; idx < 3 * CIN * XS_COLS; idx += 256) {
    const int col  = idx % XS_COLS;
    const int row  = idx / XS_COLS;    // = i*CIN + c
    const int i    = row / CIN;
    const int c    = row % CIN;
    int gcol = wo0 + col;
    if (gcol > WW_ - 1) gcol = WW_ - 1;          // clamp (masked at store)
    const int gaddr = ((b * CIN + c) * HH_ + (ho + i)) * WW_ + gcol;
    stage_g2l(x + gaddr, &xs[idx]);
  }

  // ---- weight chunk stager: chunk kc -> tap (i,j), c-range [ch*32, ch*32+32)
  auto stage_ws = [&](int kc) {
    const int tap = kc / NCPT;         // 0..8
    const int ch  = kc % NCPT;         // 0..3
    const int i   = tap / 3;
    const int j   = tap % 3;
    float* buf = &ws[(kc & 1) * WCHUNK * COUT];
#pragma unroll
    for (int e = 0; e < STAGE_OPS; ++e) {
      const int idx = e * 256 + tid;   // 0..4095
      const int cl  = idx >> 7;        // 0..31
      const int co  = idx & 127;
      const int c   = ch * WCHUNK + cl;
      const int gaddr = co * (CIN * 9) + c * 9 + i * 3 + j;
      stage_g2l(w + gaddr, &buf[idx]);
    }
  };

  v8f vmax = { -__builtin_inff(), -__builtin_inff(), -__builtin_inff(),
               -__builtin_inff(), -__builtin_inff(), -__builtin_inff(),
               -__builtin_inff(), -__builtin_inff() };
  v8f vmin = {  __builtin_inff(),  __builtin_inff(),  __builtin_inff(),
                __builtin_inff(),  __builtin_inff(),  __builtin_inff(),
                __builtin_inff(),  __builtin_inff() };
  const v8f czero = { 0.f, 0.f, 0.f, 0.f, 0.f, 0.f, 0.f, 0.f };

  v2f a;  a.y = 0.f;     // K-slots 1 (lanes 0-15) / 3 (lanes 16-31) = 0
  v2f bm; bm.y = 0.f;    // don't-care, keep 0

  stage_ws(0);                          // prologue: chunk 0 in flight

  for (int kc = 0; kc < NCHP; ++kc) {
    // Issue next chunk into the other buffer, then wait only until the
    // CURRENT chunk (and xs, issued earlier in-order) has landed.
    if (kc + 1 < NCHP) {
      stage_ws(kc + 1);
      wait_async<STAGE_OPS>();          // <=16 outstanding => chunk kc done
    } else {
      wait_async<0>();
    }
    __syncthreads();                    // chunk kc visible to all waves

    const int tap = kc / NCPT;
    const int ch  = kc % NCPT;
    const int i   = tap / 3;
    const int j   = tap % 3;
    const float* wsb = &ws[(kc & 1) * WCHUNK * COUT];
    const int xbase  = (i * CIN + ch * WCHUNK) * XS_COLS + j + lan;

#pragma unroll 4
    for (int cc = 0; cc < WCHUNK; cc += 2) {
      // A: 16 pixels at k (lanes>=16 zeroed => K-slots 2,3 dead)
      const float av0 = xs[xbase + (cc + 0) * XS_COLS];
      const float bv0 = wsb[(cc + 0) * COUT + n0 + lan];
      const float av1 = xs[xbase + (cc + 1) * XS_COLS];
      const float bv1 = wsb[(cc + 1) * COUT + n0 + lan];

      a.x  = loA ? av0 : 0.f;
      bm.x = bv0;
      v8f d0 = __builtin_amdgcn_wmma_f32_16x16x4_f32(
          false, a, false, bm, (short)0, czero, false, false);

      a.x  = loA ? av1 : 0.f;
      bm.x = bv1;
      v8f d1 = __builtin_amdgcn_wmma_f32_16x16x4_f32(
          false, a, false, bm, (short)0, czero, false, false);

      vmax = vmax8(vmax, d0);
      vmin = vmin8(vmin, d0);
      vmax = vmax8(vmax, d1);
      vmin = vmin8(vmin, d1);
    }

    __syncthreads();   // all waves done reading buf[kc&1] before re-stage
  }

  // ---- epilogue: out[b, co, ho, wo] = max + min + bias[co]
  const int co   = n0 + lan;
  const float bv = bias[co];
  const int mhi  = (lane >> 4) * 8;    // D row group per half-wave
  const int obase = ((b * COUT + co) * HO + ho) * WO;
#pragma unroll
  for (int r = 0; r < 8; ++r) {
    const int wo = wo0 + mhi + r;
    if (wo < WO) out[obase + wo] = vmax[r] + vmin[r] + bv;
  }
}

extern "C" void kernel_launch(void* const* d_in, const int* in_sizes, int n_in,
                              void* d_out, int out_size, void* d_ws,
                              size_t ws_size, hipStream_t stream) {
  const float* x    = (const float*)d_in[0];
  const float* w    = (const float*)d_in[1];
  const float* bias = (const float*)d_in[2];
  float* out        = (float*)d_out;

  const int grid = BB_ * HO * 2;       // 960 blocks: (b, ho, wo-tile)
  mamconv2d_kernel<<<grid, 256, 0, stream>>>(x, w, bias, out);
}